// TransformerGNN_20504173871296
// MI455X (gfx1250) — compile-verified
//
#include <hip/hip_runtime.h>
#include <math.h>

#define HID 256

typedef float v2f __attribute__((ext_vector_type(2)));
typedef float v8f __attribute__((ext_vector_type(8)));

// ---------------------------------------------------------------------------
// Wave32 helpers
// ---------------------------------------------------------------------------
__device__ __forceinline__ float wave_sum(float x) {
    #pragma unroll
    for (int off = 16; off >= 1; off >>= 1)
        x += __shfl_xor(x, off, 32);
    return x;
}

__device__ __forceinline__ v8f wmma_f32(v2f a, v2f b, v8f c) {
    return __builtin_amdgcn_wmma_f32_16x16x4_f32(
               false, a, false, b, (short)0, c, false, false);
}

// ---------------------------------------------------------------------------
// GEMM  C[M,N] = A[M,K] @ B[K,N] + bias[N]   via V_WMMA_F32_16X16X4_F32
// One wave computes a 32x32 C block = 2x2 WMMA tiles (register tiling:
// 2 A-frags + 2 B-frags feed 4 WMMAs per k-step -> 8 FLOP/byte loaded).
// Requires M%32==0, N%32==0, K%4==0 (true for all shapes here).
//
// Frag layouts (per ISA):
//   A 16x4 : lane (hf=lane>>4, lm=lane&15) holds A[lm][k+2hf], A[lm][k+2hf+1]
//   B 4x16 : same lane holds B[k+2hf][lm], B[k+2hf+1][lm]
//   C 16x16: lane holds rows (hf*8..hf*8+7) of column lm (8 VGPRs)
// ---------------------------------------------------------------------------
__global__ __launch_bounds__(256) void gemm_bias_wmma(
    const float* __restrict__ A, const float* __restrict__ B,
    const float* __restrict__ bias, float* __restrict__ C,
    int M, int K, int N)
{
    int wave = (blockIdx.x * blockDim.x + threadIdx.x) >> 5;
    int lane = threadIdx.x & 31;
    int nblk = N >> 5;                 // 32-wide column blocks
    int mb = wave / nblk;
    int nb = wave - mb * nblk;
    if (mb * 32 >= M) return;

    int hf = lane >> 4;
    int lm = lane & 15;
    int col0 = nb * 32 + lm;           // first B column this lane feeds
    int row0 = mb * 32 + lm;           // first A row this lane feeds

    const float* A0 = A + (size_t)row0 * K + 2 * hf;
    const float* A1 = A0 + (size_t)16 * K;
    const float* B0 = B + (size_t)(2 * hf) * N + col0;
    const float* B1 = B0 + 16;
    const size_t bstep = (size_t)4 * N;

    v8f acc00 = {}, acc01 = {}, acc10 = {}, acc11 = {};
    #pragma unroll 2
    for (int k = 0; k < K; k += 4) {
        v2f a0, a1, b0, b1;
        a0.x = A0[0]; a0.y = A0[1];
        a1.x = A1[0]; a1.y = A1[1];
        b0.x = B0[0]; b0.y = B0[N];
        b1.x = B1[0]; b1.y = B1[N];
        acc00 = wmma_f32(a0, b0, acc00);
        acc01 = wmma_f32(a0, b1, acc01);
        acc10 = wmma_f32(a1, b0, acc10);
        acc11 = wmma_f32(a1, b1, acc11);
        A0 += 4; A1 += 4; B0 += bstep; B1 += bstep;
    }

    float bb0 = bias ? bias[col0] : 0.0f;
    float bb1 = bias ? bias[col0 + 16] : 0.0f;
    int r0 = mb * 32 + hf * 8;
    float* Cr = C + (size_t)r0 * N + col0;
    #pragma unroll
    for (int r = 0; r < 8; ++r) {
        Cr[0]  = acc00[r] + bb0;
        Cr[16] = acc01[r] + bb1;
        Cr[(size_t)16 * N]      = acc10[r] + bb0;
        Cr[(size_t)16 * N + 16] = acc11[r] + bb1;
        Cr += N;
    }
}

// ---------------------------------------------------------------------------
// Encoder: tmp[n,768] = [ x@W_emb+b_emb | t@W_t+b_t | s@W_s+b_s ]
// Tiny-K GEMVs; one thread per output element.
// ---------------------------------------------------------------------------
__global__ void encode_kernel(
    const float* __restrict__ x, const float* __restrict__ t,
    const float* __restrict__ s,
    const float* __restrict__ W_emb, const float* __restrict__ b_emb,
    const float* __restrict__ W_t,   const float* __restrict__ b_t,
    const float* __restrict__ W_s,   const float* __restrict__ b_s,
    float* __restrict__ tmp, int n)
{
    int idx = blockIdx.x * blockDim.x + threadIdx.x;
    if (idx >= n * 768) return;
    int i = idx / 768;
    int c = idx - i * 768;
    float acc;
    if (c < 256) {
        acc = b_emb[c];
        const float* xr = x + (size_t)i * 36;
        #pragma unroll
        for (int j = 0; j < 36; ++j) acc += xr[j] * W_emb[j * 256 + c];
    } else if (c < 512) {
        int cc = c - 256;
        acc = b_t[cc];
        const float* tr = t + (size_t)i * 4;
        #pragma unroll
        for (int j = 0; j < 4; ++j) acc += tr[j] * W_t[j * 256 + cc];
    } else {
        int cc = c - 512;
        acc = b_s[cc];
        const float* sr = s + (size_t)i * 6;
        #pragma unroll
        for (int j = 0; j < 6; ++j) acc += sr[j] * W_s[j * 256 + cc];
    }
    tmp[idx] = acc;
}

// ---------------------------------------------------------------------------
// LayerNorm + ReLU in place, one wave32 per row of 256
// ---------------------------------------------------------------------------
__global__ __launch_bounds__(256) void ln_relu_kernel(
    float* __restrict__ h, const float* __restrict__ g,
    const float* __restrict__ b, int n)
{
    int row  = (blockIdx.x * blockDim.x + threadIdx.x) >> 5;
    int lane = threadIdx.x & 31;
    if (row >= n) return;
    float* r = h + (size_t)row * HID;
    int d0 = lane * 8;
    float v[8];
    float sum = 0.0f;
    #pragma unroll
    for (int i = 0; i < 8; ++i) { v[i] = r[d0 + i]; sum += v[i]; }
    sum = wave_sum(sum);
    float mu = sum * (1.0f / 256.0f);
    float var = 0.0f;
    #pragma unroll
    for (int i = 0; i < 8; ++i) { float d = v[i] - mu; var += d * d; }
    var = wave_sum(var) * (1.0f / 256.0f);
    float rs = rsqrtf(var + 1e-5f);
    #pragma unroll
    for (int i = 0; i < 8; ++i) {
        float o = (v[i] - mu) * rs * g[d0 + i] + b[d0 + i];
        r[d0 + i] = fmaxf(o, 0.0f);
    }
}

// ---------------------------------------------------------------------------
// CSR build (dst-sorted incoming edge lists)
// ---------------------------------------------------------------------------
__global__ void zero_ints(int* __restrict__ a, int n) {
    int i = blockIdx.x * blockDim.x + threadIdx.x;
    if (i < n) a[i] = 0;
}
__global__ void csr_count(const int* __restrict__ dst, int* __restrict__ deg, int ne) {
    int e = blockIdx.x * blockDim.x + threadIdx.x;
    if (e < ne) atomicAdd(&deg[dst[e]], 1);
}
__global__ void csr_scan(const int* __restrict__ deg, int* __restrict__ rowptr, int n) {
    if (blockIdx.x == 0 && threadIdx.x == 0) {
        int acc = 0;
        for (int i = 0; i < n; ++i) { rowptr[i] = acc; acc += deg[i]; }
        rowptr[n] = acc;
    }
}
__global__ void csr_fill(const int* __restrict__ dst, const int* __restrict__ rowptr,
                         int* __restrict__ cursor, int* __restrict__ csr, int ne) {
    int e = blockIdx.x * blockDim.x + threadIdx.x;
    if (e < ne) {
        int d = dst[e];
        int p = atomicAdd(&cursor[d], 1);
        csr[rowptr[d] + p] = e;
    }
}

// ---------------------------------------------------------------------------
// Per-node attention aggregation: one wave32 per destination node.
// Each lane owns 8 of 256 dims. Online (flash) softmax over incoming edges.
// Edge feature e = edge_attr@We + be recomputed in registers (We preloaded).
// Fused: + skip, ReLU.
// ---------------------------------------------------------------------------
__global__ __launch_bounds__(256) void node_gather(
    const float* __restrict__ q, const float* __restrict__ k,
    const float* __restrict__ v, const float* __restrict__ skip,
    const float* __restrict__ edge_attr, const int* __restrict__ src,
    const int* __restrict__ rowptr, const int* __restrict__ csr,
    const float* __restrict__ We, const float* __restrict__ be,
    float* __restrict__ hout, int n)
{
    int node = (blockIdx.x * blockDim.x + threadIdx.x) >> 5;
    int lane = threadIdx.x & 31;
    if (node >= n) return;
    int d0 = lane * 8;

    float qr[8], be8[8], wef[8][8];
    #pragma unroll
    for (int i = 0; i < 8; ++i) qr[i] = q[(size_t)node * HID + d0 + i];
    #pragma unroll
    for (int i = 0; i < 8; ++i) be8[i] = be[d0 + i];
    #pragma unroll
    for (int j = 0; j < 8; ++j)
        #pragma unroll
        for (int i = 0; i < 8; ++i) wef[j][i] = We[j * HID + d0 + i];

    float mval = -3.402823466e+38f;   // running max (segment_max identity)
    float denom = 0.0f;
    float acc[8] = {0, 0, 0, 0, 0, 0, 0, 0};

    int beg = rowptr[node], end = rowptr[node + 1];
    for (int idx = beg; idx < end; ++idx) {
        int e = csr[idx];
        int s = src[e];
        const float* ea = edge_attr + (size_t)e * 8;
        float ef[8];
        #pragma unroll
        for (int i = 0; i < 8; ++i) ef[i] = be8[i];
        #pragma unroll
        for (int j = 0; j < 8; ++j) {
            float aj = ea[j];
            #pragma unroll
            for (int i = 0; i < 8; ++i) ef[i] += aj * wef[j][i];
        }
        const float* krow = k + (size_t)s * HID + d0;
        float partial = 0.0f;
        #pragma unroll
        for (int i = 0; i < 8; ++i) partial += qr[i] * (krow[i] + ef[i]);
        partial = wave_sum(partial);
        float alpha = partial * 0.0625f;             // 1/sqrt(256)

        float nm = fmaxf(mval, alpha);
        float scale = __expf(mval - nm);
        float w = __expf(alpha - nm);
        denom = denom * scale + w;
        const float* vrow = v + (size_t)s * HID + d0;
        #pragma unroll
        for (int i = 0; i < 8; ++i) acc[i] = acc[i] * scale + w * (vrow[i] + ef[i]);
        mval = nm;
    }

    float inv = 1.0f / (denom + 1e-16f);
    const float* sr = skip + (size_t)node * HID + d0;
    float* outr = hout + (size_t)node * HID + d0;
    #pragma unroll
    for (int i = 0; i < 8; ++i)
        outr[i] = fmaxf(acc[i] * inv + sr[i], 0.0f);
}

// ---------------------------------------------------------------------------
// Decoder: out[n,18] = h @ W_dec[256,18] + b_dec
// ---------------------------------------------------------------------------
__global__ void decoder_kernel(const float* __restrict__ h,
                               const float* __restrict__ Wd,
                               const float* __restrict__ bd,
                               float* __restrict__ out, int n)
{
    int idx = blockIdx.x * blockDim.x + threadIdx.x;
    if (idx >= n * 18) return;
    int i = idx / 18;
    int o = idx - i * 18;
    float acc = bd[o];
    const float* hr = h + (size_t)i * HID;
    for (int c = 0; c < HID; ++c) acc += hr[c] * Wd[c * 18 + o];
    out[idx] = acc;
}

// ---------------------------------------------------------------------------
extern "C" void kernel_launch(void* const* d_in, const int* in_sizes, int n_in,
                              void* d_out, int out_size, void* d_ws, size_t ws_size,
                              hipStream_t stream)
{
    const float* x     = (const float*)d_in[0];
    const int*   eidx  = (const int*)  d_in[1];
    const float* eattr = (const float*)d_in[2];
    const float* t     = (const float*)d_in[3];
    const float* s     = (const float*)d_in[4];
    const float* W_emb = (const float*)d_in[5];
    const float* b_emb = (const float*)d_in[6];
    const float* W_t   = (const float*)d_in[7];
    const float* b_t   = (const float*)d_in[8];
    const float* W_s   = (const float*)d_in[9];
    const float* b_s   = (const float*)d_in[10];
    const float* W_f   = (const float*)d_in[11];
    const float* b_f   = (const float*)d_in[12];
    const float* ln_g  = (const float*)d_in[13];
    const float* ln_b  = (const float*)d_in[14];
    const float* Wq    = (const float*)d_in[15];
    const float* bq    = (const float*)d_in[16];
    const float* Wk    = (const float*)d_in[17];
    const float* bk    = (const float*)d_in[18];
    const float* Wv    = (const float*)d_in[19];
    const float* bv    = (const float*)d_in[20];
    const float* We    = (const float*)d_in[21];
    const float* be    = (const float*)d_in[22];
    const float* Wsk   = (const float*)d_in[23];
    const float* bsk   = (const float*)d_in[24];
    const float* Wd    = (const float*)d_in[25];
    const float* bd    = (const float*)d_in[26];

    const int n  = in_sizes[0] / 36;   // 20000
    const int ne = in_sizes[1] / 2;    // 640000
    const int* src = eidx;
    const int* dst = eidx + ne;

    // Workspace layout (all fits in L2 on MI455X; ~126 MB total)
    float* wsf  = (float*)d_ws;
    float* h_a  = wsf;
    float* h_b  = h_a + (size_t)n * HID;
    float* qb   = h_b + (size_t)n * HID;
    float* kb   = qb  + (size_t)n * HID;
    float* vb   = kb  + (size_t)n * HID;
    float* skb  = vb  + (size_t)n * HID;
    float* tmp  = qb;                       // n*768, aliases q|k|v (disjoint in time)
    int* deg    = (int*)(skb + (size_t)n * HID);
    int* rowptr = deg + n;
    int* cursor = rowptr + n + 1;
    int* csr    = cursor + n;

    auto gemm = [&](const float* A, const float* B, const float* bias, float* C,
                    int M, int K, int N) {
        int waves  = ((M + 31) / 32) * (N / 32);   // one wave per 32x32 C block
        int blocks = (waves + 7) / 8;
        gemm_bias_wmma<<<blocks, 256, 0, stream>>>(A, B, bias, C, M, K, N);
    };

    // ---- CSR build (edge_index is constant; rebuilt deterministically) ----
    zero_ints<<<(n + 255) / 256, 256, 0, stream>>>(deg, n);
    zero_ints<<<(n + 255) / 256, 256, 0, stream>>>(cursor, n);
    csr_count<<<(ne + 255) / 256, 256, 0, stream>>>(dst, deg, ne);
    csr_scan<<<1, 1, 0, stream>>>(deg, rowptr, n);
    csr_fill<<<(ne + 255) / 256, 256, 0, stream>>>(dst, rowptr, cursor, csr, ne);

    // ---- Encoder: tmp[n,768] -> fuse GEMM -> LN+ReLU -> h_a ----
    encode_kernel<<<(n * 768 + 255) / 256, 256, 0, stream>>>(
        x, t, s, W_emb, b_emb, W_t, b_t, W_s, b_s, tmp, n);
    gemm(tmp, W_f, b_f, h_a, n, 768, HID);
    ln_relu_kernel<<<(n + 7) / 8, 256, 0, stream>>>(h_a, ln_g, ln_b, n);

    // ---- 5 TransformerConv layers ----
    for (int i = 0; i < 5; ++i) {
        const float* hc = (i & 1) ? h_b : h_a;
        float*       hn = (i & 1) ? h_a : h_b;
        const size_t wo = (size_t)i * HID * HID;
        gemm(hc, Wq  + wo, bq  + i * HID, qb,  n, HID, HID);
        gemm(hc, Wk  + wo, bk  + i * HID, kb,  n, HID, HID);
        gemm(hc, Wv  + wo, bv  + i * HID, vb,  n, HID, HID);
        gemm(hc, Wsk + wo, bsk + i * HID, skb, n, HID, HID);
        node_gather<<<(n + 7) / 8, 256, 0, stream>>>(
            qb, kb, vb, skb, eattr, src, rowptr, csr,
            We + (size_t)i * 8 * HID, be + i * HID, hn, n);
    }

    // ---- Decoder (final h is in h_b after 5 layers) ----
    decoder_kernel<<<(n * 18 + 255) / 256, 256, 0, stream>>>(
        h_b, Wd, bd, (float*)d_out, n);
}